// Gauss_agg1_st_50208167690568
// MI455X (gfx1250) — compile-verified
//
#include <hip/hip_runtime.h>
#include <stdint.h>

typedef __attribute__((ext_vector_type(2))) float    v2f;
typedef __attribute__((ext_vector_type(8))) float    v8f;
typedef __attribute__((ext_vector_type(4))) unsigned v4u;
typedef __attribute__((ext_vector_type(4))) int      v4i;
typedef __attribute__((ext_vector_type(8))) int      v8i;

#define T_ALL 600
#define ROWF  15      // floats per time row: 5 joints * 3 channels
#define TILE  100     // output positions per workgroup (divides all window lengths)
#define MAXR  102     // max LDS rows: TILE + 2 halo

__device__ __forceinline__ int imax(int a, int b) { return a > b ? a : b; }
__device__ __forceinline__ int imin(int a, int b) { return a < b ? a : b; }

__global__ __launch_bounds__(256)
void gauss_agg1_cdna5_kernel(const float* __restrict__ x, float* __restrict__ out)
{
    __shared__ float tile[MAXR * ROWF];      // staged input rows (packed, stride 15 floats)
    __shared__ float muS[TILE][3];           // per-position weighted means

    const int tau = (int)blockIdx.x;         // tile index 0..17
    const int n   = (int)blockIdx.y;         // joint group 0..4
    const int b   = (int)blockIdx.z;         // batch 0..255

    // window decode: lengths 600,300,300,200,200,200 ; abs bases 0,0,300,0,200,400
    int w, l0;
    if      (tau < 6)  { w = 0; l0 = (tau     ) * TILE; }
    else if (tau < 9)  { w = 1; l0 = (tau -  6) * TILE; }
    else if (tau < 12) { w = 2; l0 = (tau -  9) * TILE; }
    else if (tau < 14) { w = 3; l0 = (tau - 12) * TILE; }
    else if (tau < 16) { w = 4; l0 = (tau - 14) * TILE; }
    else               { w = 5; l0 = (tau - 16) * TILE; }
    const int wBase = (w == 2) ? 300 : (w == 4) ? 200 : (w == 5) ? 400 : 0;
    const int wLen  = (w == 0) ? 600 : (w < 3) ? 300 : 200;
    const int wOut  = (w == 0) ? 0 : (w == 1) ? 600 : (w == 2) ? 900
                    : (w == 3) ? 1200 : (w == 4) ? 1400 : 1600;

    const int t0       = wBase + l0;                 // first absolute time of tile
    const int rowStart = imax(t0 - 1, 0);            // halo row below
    const int rowEnd   = imin(t0 + TILE, T_ALL - 1); // halo row above
    const int numRows  = rowEnd - rowStart + 1;      // <= 102

    // ---- TDM: DMA the (numRows x 15 float) tile, row stride 75 floats, into LDS ----
    if (threadIdx.x < 32) {
        unsigned long long gaddr = (unsigned long long)(uintptr_t)x +
            4ull * ((unsigned long long)(b * T_ALL + rowStart) * 25ull +
                    (unsigned long long)(n * 5)) * 3ull;
        unsigned ldsOff = (unsigned)(uintptr_t)(&tile[0]);

        v4u g0; v8i g1; v4i g2, g3; v8i g4;
        g0[0] = 1u;                                           // count=1, load, user mode
        g0[1] = ldsOff;                                       // lds_addr
        g0[2] = (unsigned)(gaddr & 0xffffffffull);            // global_addr[31:0]
        g0[3] = ((unsigned)((gaddr >> 32) & 0x1ffffffull))    // global_addr[56:32]
              | 0x80000000u;                                  // type=2 ("image")
        g1[0] = (int)(2u << 16);                              // data_size=2 -> 4 bytes
        g1[1] = (int)(((unsigned)ROWF) << 16);                // tensor_dim0 = 15
        g1[2] = (int)(((unsigned)numRows) << 16);             // tensor_dim1[15:0]
        g1[3] = (int)(((unsigned)ROWF) << 16);                // tile_dim0 = 15
        g1[4] = numRows;                                      // tile_dim1
        g1[5] = 75;                                           // tensor_dim0_stride (floats)
        g1[6] = 0; g1[7] = 0;
        g2[0] = g2[1] = g2[2] = g2[3] = 0;
        g3[0] = g3[1] = g3[2] = g3[3] = 0;
        g4[0] = g4[1] = g4[2] = g4[3] = 0;
        g4[4] = g4[5] = g4[6] = g4[7] = 0;

        __builtin_amdgcn_tensor_load_to_lds(g0, g1, g2, g3, g4, 0);
        __builtin_amdgcn_s_wait_tensorcnt(0);
    }
    __syncthreads();

    // ---- per-position weighted mean mu = (4,2,3,2,4)/15 over the 5 joints ----
    if (threadIdx.x < TILE) {
        const int row = (t0 + (int)threadIdx.x) - rowStart;
        const float* r = &tile[row * ROWF];
        #pragma unroll
        for (int c = 0; c < 3; ++c) {
            muS[threadIdx.x][c] =
                (4.f * r[c] + 2.f * r[3 + c] + 3.f * r[6 + c] +
                 2.f * r[9 + c] + 4.f * r[12 + c]) * (1.0f / 15.0f);
        }
    }
    __syncthreads();

    // ---- WMMA phase: 4 output positions per 16x16 Gram tile, K=16 via 4 chained WMMAs ----
    const int lane = (int)(threadIdx.x & 31);
    const int wv   = (int)(threadIdx.x >> 5);
    const int half = lane >> 4;       // selects K pair (0,1) vs (2,3)
    const int lid  = lane & 15;       // M (A) / N (B) index
    const int pos  = lid >> 2;        // which of 4 packed positions
    const int cd   = lid & 3;         // channel index (c for A row, d for B col)
    const float S15  = 3.872983346207417f;   // sqrt(15)
    const float S3   = 1.7320508075688772f;  // sqrt(3)
    const float inv15 = 1.0f / 15.0f;

    for (int g = wv; g < 25; g += 8) {
        const int lpos = 4 * g + pos;         // local position 0..99
        const int t    = t0 + lpos;           // absolute time
        const int lw   = t - wBase;           // position within window
        const bool isB = (lw == 0) || (lw == wLen - 1);
        const float mu0 = muS[lpos][0], mu1 = muS[lpos][1], mu2 = muS[lpos][2];
        const float muc = (cd == 0) ? mu0 : (cd == 1) ? mu1 : (cd == 2) ? mu2 : 0.0f;
        const int rbase = (t - rowStart) * ROWF;

        v8f acc = {};
        #pragma unroll
        for (int chunk = 0; chunk < 4; ++chunk) {
            v2f ab;
            #pragma unroll
            for (int vg = 0; vg < 2; ++vg) {
                const int k  = chunk * 4 + half * 2 + vg;   // contraction row 0..15
                const int dt = k / 5 - 1;                   // -1,0,+1 (junk for k=15)
                const int jj = k - (dt + 1) * 5;            // joint 0..4
                int ridx = rbase + (isB ? 0 : dt * ROWF) + jj * 3 + cd;
                ridx = imin(imax(ridx, 0), MAXR * ROWF - 1);
                const float y = tile[ridx] - muc;
                float val;
                if (k == 15)      val = (cd == 3) ? S15 : S15 * muc;  // sqrt(15)*[mu,1]
                else if (cd == 3) val = 0.0f;
                else if (isB)     val = (dt == 0) ? S3 * y : 0.0f;    // boundary: sqrt(3)*(x-mu)
                else              val = y;                            // interior: x[t+dt]-mu[t]
                ab[vg] = val;
            }
            // D = A(16x4) x B(4x16) + C ; symmetric construction -> same regs for A and B
            acc = __builtin_amdgcn_wmma_f32_16x16x4_f32(
                false, ab, false, ab, (short)0, acc, false, false);
        }

        // Diagonal 4x4 blocks of D are the augmented Gaussians; output is symmetric,
        // so each useful lane (column d of a block) stores row d as one float4.
        if (lane < 8 || lane >= 24) {
            const int nc = (lane < 8) ? lane : (lane - 16);  // N column 0..15
            const int p2 = nc >> 2;                          // packed position 0..3
            const int d  = nc & 3;                           // row of 4x4 to store
            const bool hi = (p2 & 1) != 0;                   // acc VGPR bank select
            float4 v;
            v.x = (hi ? acc[4] : acc[0]) * inv15;
            v.y = (hi ? acc[5] : acc[1]) * inv15;
            v.z = (hi ? acc[6] : acc[2]) * inv15;
            v.w = (hi ? acc[7] : acc[3]) * inv15;
            const size_t outPos = (size_t)(wOut + l0 + 4 * g + p2);
            const size_t oidx = ((((size_t)b * 5 + (size_t)n) * 1800 + outPos) * 16)
                              + (size_t)(d * 4);
            *reinterpret_cast<float4*>(out + oidx) = v;
        }
    }
}

extern "C" void kernel_launch(void* const* d_in, const int* in_sizes, int n_in,
                              void* d_out, int out_size, void* d_ws, size_t ws_size,
                              hipStream_t stream) {
    (void)in_sizes; (void)n_in; (void)out_size; (void)d_ws; (void)ws_size;
    const float* x = (const float*)d_in[0];
    float* out = (float*)d_out;
    dim3 grid(18, 5, 256);   // (tile, joint-group, batch)
    gauss_agg1_cdna5_kernel<<<grid, dim3(256), 0, stream>>>(x, out);
}